// IoE_27779848470735
// MI455X (gfx1250) — compile-verified
//
#include <hip/hip_runtime.h>

// Problem geometry: x is [1, D, H, W] = [1, 512, 512, 64] float32, C-contiguous.
#define D_DIM 512
#define H_DIM 512
#define W_DIM 64
#define TH    16   // h-rows per block

// ---------------------------------------------------------------------------
// CDNA5 async-to-LDS staging path (gfx1250). The builtin takes pointers to
// 4 x i32 vectors: (v4i AS1*, v4i AS3*, imm offset, imm cpol).
// Falls back to a synchronous load+ds_store copy if unavailable.
// ---------------------------------------------------------------------------
#if defined(__gfx1250__) && __has_builtin(__builtin_amdgcn_global_load_async_to_lds_b128)
#define HAS_ASYNC_LDS 1
#else
#define HAS_ASYNC_LDS 0
#endif

#if HAS_ASYNC_LDS
typedef int v4i_t __attribute__((vector_size(16)));
typedef __attribute__((address_space(1))) v4i_t gas_v4i;
typedef __attribute__((address_space(3))) v4i_t las_v4i;
#endif

__device__ __forceinline__ void stage_b128(const float* gsrc, float* ldst) {
#if HAS_ASYNC_LDS
    __builtin_amdgcn_global_load_async_to_lds_b128(
        (gas_v4i*)gsrc, (las_v4i*)ldst, /*offset=*/0, /*cpol=*/0);
#else
    *(float4*)ldst = *(const float4*)gsrc;
#endif
}

__device__ __forceinline__ void wait_async_all() {
#if HAS_ASYNC_LDS
#if __has_builtin(__builtin_amdgcn_s_wait_asynccnt)
    __builtin_amdgcn_s_wait_asynccnt(0);
#else
    asm volatile("s_wait_asynccnt 0" ::: "memory");
#endif
#endif
}

// jnp.where(t != 0, 1.0, t): keeps +0/-0 as-is, 1.0 otherwise.
__device__ __forceinline__ float stepnz(float t) {
    return (t != 0.0f) ? 1.0f : t;
}

extern "C" __global__ void __launch_bounds__(256)
IoE_fd3_stencil(const float* __restrict__ x, float* __restrict__ out) {
    // Tile rows h0-1 .. h0+TH (18 rows of 64 floats, 4.5 KB) for one d-slice.
    __shared__ __align__(16) float tile[(TH + 2) * W_DIM];

    const int    tid   = threadIdx.x;
    const int    blk   = (int)blockIdx.x;
    const int    d     = blk >> 5;          // 512 d-slices
    const int    h0    = (blk & 31) * TH;   // 32 h-tiles of 16 rows
    const size_t slice = (size_t)H_DIM * W_DIM;   // 32768 floats = 128 KiB
    const float* xd    = x + (size_t)d * slice;

    // ---- Stage center slab rows [h0-1, h0+TH] into LDS (async B128). ----
    // 18 rows * 16 float4 = 288 transfers over 256 threads.
    for (int i = tid; i < (TH + 2) * (W_DIM / 4); i += 256) {
        const int row = i >> 4;             // 0..17
        const int c   = (i & 15) << 2;      // 0,4,...,60
        const int gh  = h0 - 1 + row;       // global h of this row
        float*    dst = &tile[row * W_DIM + c];
        if ((unsigned)gh < (unsigned)H_DIM) {
            stage_b128(xd + (size_t)gh * W_DIM + c, dst);
        } else {
            // zero-padding halo outside the H range
            dst[0] = 0.0f; dst[1] = 0.0f; dst[2] = 0.0f; dst[3] = 0.0f;
        }
    }
    wait_async_all();
    __syncthreads();

    // ---- Each thread produces one float4 along W. ----
    const int lane = tid & 15;              // which float4 chunk of the row
    const int hs   = tid >> 4;              // 0..15 within the h-tile
    const int w    = lane << 2;             // element offset in row
    const int h    = h0 + hs;

    const float* rc = &tile[(hs + 1) * W_DIM];       // center row (global h)
    const float4 c  = *(const float4*)(rc + w);
    // W halo: zero-padded at w==-1 / w==64. Reads below stay inside `tile`
    // (they touch adjacent tile rows in the OOB-lane case and are masked out).
    const float left  = (lane == 0)  ? 0.0f : rc[w - 1];
    const float right = (lane == 15) ? 0.0f : rc[w + 4];

    const float4 hu = *(const float4*)(&tile[hs       * W_DIM + w]); // h-1
    const float4 hd = *(const float4*)(&tile[(hs + 2) * W_DIM + w]); // h+1

    // D neighbors: +/-128 KiB strides; entire 64 MiB input is L2-resident.
    const size_t base = (size_t)d * slice + (size_t)h * W_DIM + (size_t)w;
    float4 du = make_float4(0.0f, 0.0f, 0.0f, 0.0f);
    float4 dd = make_float4(0.0f, 0.0f, 0.0f, 0.0f);
    if (d > 0)         du = *(const float4*)(x + (base - slice));
    if (d < D_DIM - 1) dd = *(const float4*)(x + (base + slice));

    float4 r;
    r.x = stepnz(c.y   - left) + stepnz(hd.x - hu.x) + stepnz(dd.x - du.x);
    r.y = stepnz(c.z   - c.x)  + stepnz(hd.y - hu.y) + stepnz(dd.y - du.y);
    r.z = stepnz(c.w   - c.y)  + stepnz(hd.z - hu.z) + stepnz(dd.z - du.z);
    r.w = stepnz(right - c.z)  + stepnz(hd.w - hu.w) + stepnz(dd.w - du.w);

    *(float4*)(out + base) = r;
}

extern "C" void kernel_launch(void* const* d_in, const int* in_sizes, int n_in,
                              void* d_out, int out_size, void* d_ws, size_t ws_size,
                              hipStream_t stream) {
    (void)in_sizes; (void)n_in; (void)out_size; (void)d_ws; (void)ws_size;
    const float* x   = (const float*)d_in[0];
    float*       out = (float*)d_out;
    // 512 d-slices * 32 h-tiles = 16384 blocks of 256 threads (8 wave32s).
    dim3 grid(D_DIM * (H_DIM / TH));
    IoE_fd3_stencil<<<grid, 256, 0, stream>>>(x, out);
}